// _SAmodule_57019985822402
// MI455X (gfx1250) — compile-verified
//
#include <hip/hip_runtime.h>
#include <hip/hip_bf16.h>

// Shapes: B=1, C=8, D=H=W=32. HW=1024, DHW=32768, CDHW=262144.
// Workspace layout (floats): q[262144] k[262144] v[262144] Sinv[1048576] Sdinv[1024]
//   total ~7.3 MB (L2 is 192 MB -> everything stays L2-resident; HBM traffic ~10 MB one-shot).
// Cost model: attention core = ~0.55G v_exp_f32 + ~1.6 GFLOP VALU (no matmul structure:
// exp of an elementwise outer product cannot be factored into GEMM). WMMA is used where
// matmul structure exists: the 1x1x1 and 3x3x3 convs, in exact f32 (V_WMMA_F32_16X16X4_F32).

typedef __attribute__((ext_vector_type(2))) float v2f;
typedef __attribute__((ext_vector_type(8))) float v8f;

#define CH   8
#define DIM  32
#define HW   1024
#define DHW  32768
#define CDHW 262144

static __device__ __forceinline__ int clamp31(int a) {
    return a < 0 ? 0 : (a > 31 ? 31 : a);
}

// ---------------------------------------------------------------------------
// K1: q/k (1x1x1 convs) + v (3x3x3 conv SAME) via V_WMMA_F32_16X16X4_F32.
// One wave per tile of 16 consecutive voxels (half a W-row).
//  - q&k fused in one accumulator: A rows 0-7 = w_plane, rows 8-15 = w_depth.
//  - v: K-dim = 27 taps * 8 ic = 216 -> 54 WMMAs of K=4.
// Boundary handling: clamped (always in-range) loads + select-to-zero, so EXEC
// stays all-ones throughout (WMMA requirement) and no OOB address is formed.
// ---------------------------------------------------------------------------
__global__ __launch_bounds__(256) void qkv_wmma(
    const float* __restrict__ x,
    const float* __restrict__ wp, const float* __restrict__ bp,
    const float* __restrict__ wd, const float* __restrict__ bd,
    const float* __restrict__ wb, const float* __restrict__ bb,
    float* __restrict__ q, float* __restrict__ k, float* __restrict__ v)
{
    const int lane = threadIdx.x & 31;
    const int wave = threadIdx.x >> 5;
    const int tile = blockIdx.x * 8 + wave;          // 2048 tiles of 16 voxels
    const int p0   = tile * 16;
    const int d    = p0 >> 10;
    const int h    = (p0 >> 5) & 31;
    const int w0   = p0 & 31;                        // 0 or 16

    const int M  = lane & 15;                        // A-matrix row for this lane
    const int kb = (lane >> 4) << 1;                 // K base: 0 (lanes 0-15) or 2
    const int N  = lane & 15;                        // B/D column for this lane
    const int p  = p0 + N;

    // ---- q (rows 0-7) and k (rows 8-15) fused in one accumulator ----
    v8f accqk = {};
    #pragma unroll
    for (int ch = 0; ch < 2; ++ch) {
        const int c0 = ch * 4;
        const float* wA = (M < 8) ? (wp + M * CH) : (wd + (M - 8) * CH);
        v2f a, b;
        a.x = wA[c0 + kb];
        a.y = wA[c0 + kb + 1];
        b.x = x[(c0 + kb)     * DHW + p];
        b.y = x[(c0 + kb + 1) * DHW + p];
        accqk = __builtin_amdgcn_wmma_f32_16x16x4_f32(
            false, a, false, b, (short)0, accqk, false, false);
    }
    {
        float*       base = (lane < 16) ? q  : k;    // lanes 0-15: q rows, 16-31: k rows
        const float* bias = (lane < 16) ? bp : bd;
        #pragma unroll
        for (int r = 0; r < 8; ++r)
            base[r * DHW + p] = accqk[r] + bias[r];
    }

    // ---- v: 3x3x3 conv, 54 K=4 chunks (tap = ck/2, ic base = (ck&1)*4) ----
    v8f accv = {};
    for (int ck = 0; ck < 54; ++ck) {
        const int tap = ck >> 1;
        const int icb = (ck & 1) * 4;
        const int dz  = tap / 9 - 1;
        const int dy  = (tap / 3) % 3 - 1;
        const int dx  = tap % 3 - 1;

        v2f a, b;
        {   // rows 8-15 of A are zero
            const int wi = M * 216 + (icb + kb) * 27 + tap;   // (o,ic,kd,kh,kw)
            const bool arow = (M < 8);
            const float a0 = wb[arow ? wi      : 0];
            const float a1 = wb[arow ? wi + 27 : 0];
            a.x = arow ? a0 : 0.0f;
            a.y = arow ? a1 : 0.0f;
        }

        const int dd = d + dz, hh = h + dy, ww = w0 + N + dx;
        const bool inb = ((unsigned)dd < 32u) & ((unsigned)hh < 32u) & ((unsigned)ww < 32u);
        const int  si  = clamp31(dd) * HW + clamp31(hh) * 32 + clamp31(ww);
        const float b0 = x[(icb + kb)     * DHW + si];       // always in-range
        const float b1 = x[(icb + kb + 1) * DHW + si];
        b.x = inb ? b0 : 0.0f;                               // v_cndmask, EXEC untouched
        b.y = inb ? b1 : 0.0f;

        accv = __builtin_amdgcn_wmma_f32_16x16x4_f32(
            false, a, false, b, (short)0, accv, false, false);
    }
    if (lane < 16) {                                  // only rows 0-7 are valid v channels
        #pragma unroll
        for (int r = 0; r < 8; ++r)
            v[r * DHW + p0 + lane] = accv[r] + bb[r];
    }
}

// ---------------------------------------------------------------------------
// K2: plane softmax denominator. Sinv[i,j] = 1 / sum_m exp(q[m,i]*k[m,j]).
// 4 j's per thread: per m-iter = 1 wave-uniform q load + 1 b128 k load.
// i == blockIdx.x (uniform per block); j coverage contiguous & 16B-aligned.
// ---------------------------------------------------------------------------
__global__ __launch_bounds__(256) void plane_denom(
    const float* __restrict__ q, const float* __restrict__ k,
    float* __restrict__ Sinv)
{
    const int i  = blockIdx.x;                       // 1024 blocks
    const int j4 = threadIdx.x << 2;                 // 0..1020
    const float* qi = q + i;
    float4 s = make_float4(0.f, 0.f, 0.f, 0.f);
    #pragma unroll 2
    for (int m = 0; m < 256; ++m) {
        const float  qm = qi[m * HW];
        const float4 km = *(const float4*)(k + m * HW + j4);
        s.x += __expf(qm * km.x);
        s.y += __expf(qm * km.y);
        s.z += __expf(qm * km.z);
        s.w += __expf(qm * km.w);
    }
    float4 r;
    r.x = 1.0f / s.x; r.y = 1.0f / s.y; r.z = 1.0f / s.z; r.w = 1.0f / s.w;
    *(float4*)(Sinv + i * HW + j4) = r;
}

// ---------------------------------------------------------------------------
// K3: p_att fused with output:
//   out[n,j] = x[n,j] + g * sum_i exp(q[n,i]k[n,j]) * v[n,i] * Sinv[i,j]
// 4 n's per thread amortize the Sinv stream; i unrolled by 4 with b128 q/v loads.
// ---------------------------------------------------------------------------
__global__ __launch_bounds__(256) void plane_att(
    const float* __restrict__ x, const float* __restrict__ q,
    const float* __restrict__ k, const float* __restrict__ v,
    const float* __restrict__ Sinv, const float* __restrict__ gamma,
    float* __restrict__ out)
{
    const int t  = blockIdx.x * 256 + threadIdx.x;   // 65536 threads
    const int j  = t & 1023;
    const int n0 = (t >> 10) * 4;

    const float k0 = k[(n0 + 0) * HW + j];
    const float k1 = k[(n0 + 1) * HW + j];
    const float k2 = k[(n0 + 2) * HW + j];
    const float k3 = k[(n0 + 3) * HW + j];
    float a0 = 0.f, a1 = 0.f, a2 = 0.f, a3 = 0.f;

    for (int i = 0; i < HW; i += 4) {
        const float s0 = Sinv[(i + 0) * HW + j];     // coalesced b32 per row
        const float s1 = Sinv[(i + 1) * HW + j];
        const float s2 = Sinv[(i + 2) * HW + j];
        const float s3 = Sinv[(i + 3) * HW + j];
        const float4 q0 = *(const float4*)(q + (n0 + 0) * HW + i);
        const float4 q1 = *(const float4*)(q + (n0 + 1) * HW + i);
        const float4 q2 = *(const float4*)(q + (n0 + 2) * HW + i);
        const float4 q3 = *(const float4*)(q + (n0 + 3) * HW + i);
        const float4 v0 = *(const float4*)(v + (n0 + 0) * HW + i);
        const float4 v1 = *(const float4*)(v + (n0 + 1) * HW + i);
        const float4 v2 = *(const float4*)(v + (n0 + 2) * HW + i);
        const float4 v3 = *(const float4*)(v + (n0 + 3) * HW + i);

        a0 += __expf(q0.x * k0) * v0.x * s0;  a0 += __expf(q0.y * k0) * v0.y * s1;
        a0 += __expf(q0.z * k0) * v0.z * s2;  a0 += __expf(q0.w * k0) * v0.w * s3;
        a1 += __expf(q1.x * k1) * v1.x * s0;  a1 += __expf(q1.y * k1) * v1.y * s1;
        a1 += __expf(q1.z * k1) * v1.z * s2;  a1 += __expf(q1.w * k1) * v1.w * s3;
        a2 += __expf(q2.x * k2) * v2.x * s0;  a2 += __expf(q2.y * k2) * v2.y * s1;
        a2 += __expf(q2.z * k2) * v2.z * s2;  a2 += __expf(q2.w * k2) * v2.w * s3;
        a3 += __expf(q3.x * k3) * v3.x * s0;  a3 += __expf(q3.y * k3) * v3.y * s1;
        a3 += __expf(q3.z * k3) * v3.z * s2;  a3 += __expf(q3.w * k3) * v3.w * s3;
    }
    const float g = gamma[0];
    out[(n0 + 0) * HW + j] = x[(n0 + 0) * HW + j] + g * a0;
    out[(n0 + 1) * HW + j] = x[(n0 + 1) * HW + j] + g * a1;
    out[(n0 + 2) * HW + j] = x[(n0 + 2) * HW + j] + g * a2;
    out[(n0 + 3) * HW + j] = x[(n0 + 3) * HW + j] + g * a3;
}

// ---------------------------------------------------------------------------
// K4: depth softmax denominator. Sdinv[a,b] = 1 / sum_{n2<8192} exp(qd[n2,a]*kd[n2,b])
// One 256-thread block per (a,b); coalesced hw-major loads; tree reduction.
// ---------------------------------------------------------------------------
__global__ __launch_bounds__(256) void depth_denom(
    const float* __restrict__ q, const float* __restrict__ k,
    float* __restrict__ Sdinv)
{
    const int ab = blockIdx.x;                       // 1024 blocks
    const int a  = ab >> 5, b = ab & 31;
    float part = 0.0f;
    for (int n2 = threadIdx.x; n2 < 8192; n2 += 256) {
        const int c = n2 >> 10, hw = n2 & 1023;
        part += __expf(q[c * DHW + a * HW + hw] * k[c * DHW + b * HW + hw]);
    }
    #pragma unroll
    for (int off = 16; off > 0; off >>= 1)
        part += __shfl_down(part, off, 32);
    __shared__ float red[8];
    if ((threadIdx.x & 31) == 0) red[threadIdx.x >> 5] = part;
    __syncthreads();
    if (threadIdx.x == 0) {
        float s = 0.0f;
        #pragma unroll
        for (int w = 0; w < 8; ++w) s += red[w];
        Sdinv[ab] = 1.0f / s;
    }
}

// ---------------------------------------------------------------------------
// K5: depth attention + legacy transpose(2,4) scatter, accumulated into out.
// dsum[n2,b] = sum_a exp(qd[n2,a]kd[n2,b])*vd[n2,a]*Sdinv[a,b];
// out[c, b, w, h] += g * dsum[c*1024 + h*32 + w, b]   (n2 = c*1024+h*32+w)
// ---------------------------------------------------------------------------
__global__ __launch_bounds__(256) void depth_att(
    const float* __restrict__ q, const float* __restrict__ k,
    const float* __restrict__ v, const float* __restrict__ Sdinv,
    const float* __restrict__ gamma, float* __restrict__ out)
{
    const int t  = blockIdx.x * 256 + threadIdx.x;   // 262144 threads
    const int b  = t & 31;
    const int n2 = t >> 5;
    const int c  = n2 >> 10, hw = n2 & 1023;

    const float kb = k[c * DHW + b * HW + hw];
    float acc = 0.0f;
    #pragma unroll 4
    for (int a = 0; a < DIM; ++a) {
        const float qa = q[c * DHW + a * HW + hw];   // wave-uniform (broadcast)
        const float va = v[c * DHW + a * HW + hw];
        acc += __expf(qa * kb) * va * Sdinv[a * 32 + b];
    }
    const int h  = (n2 >> 5) & 31;
    const int w  = n2 & 31;
    const int oi = c * DHW + b * HW + w * 32 + h;    // swapaxes(2,4) of raw buffer
    out[oi] += gamma[0] * acc;
}

// ---------------------------------------------------------------------------
extern "C" void kernel_launch(void* const* d_in, const int* in_sizes, int n_in,
                              void* d_out, int out_size, void* d_ws, size_t ws_size,
                              hipStream_t stream) {
    const float* x     = (const float*)d_in[0];
    const float* wp    = (const float*)d_in[1];
    const float* bp    = (const float*)d_in[2];
    const float* wd    = (const float*)d_in[3];
    const float* bd    = (const float*)d_in[4];
    const float* wb    = (const float*)d_in[5];
    const float* bb    = (const float*)d_in[6];
    const float* gamma = (const float*)d_in[7];
    float* out = (float*)d_out;

    float* q     = (float*)d_ws;           // 262144
    float* k     = q + CDHW;               // 262144
    float* v     = k + CDHW;               // 262144
    float* Sinv  = v + CDHW;               // 1048576
    float* Sdinv = Sinv + HW * HW;         // 1024

    qkv_wmma   <<<256,  256, 0, stream>>>(x, wp, bp, wd, bd, wb, bb, q, k, v);
    plane_denom<<<1024, 256, 0, stream>>>(q, k, Sinv);
    plane_att  <<<256,  256, 0, stream>>>(x, q, k, v, Sinv, gamma, out);
    depth_denom<<<1024, 256, 0, stream>>>(q, k, Sdinv);
    depth_att  <<<1024, 256, 0, stream>>>(q, k, v, Sdinv, gamma, out);
}